// AsymmetricContrastiveLoss_82660940579426
// MI455X (gfx1250) — compile-verified
//
#include <hip/hip_runtime.h>

#define NPOS   16384
#define DIM    2048
#define TD     512
#define EPSF   1e-8f
#define CHUNK  64          // K-chunk (floats per slot) staged in LDS
#define LSTR   68          // padded LDS slot stride in floats (16B aligned, bank-conflict free)
#define K1_WAVES 8
#define K2_WAVES 8
#define K1_BLOCKS (NPOS / 4 / K1_WAVES)    // 512
#define K2_BLOCKS (NPOS / 16 / K2_WAVES)   // 128

typedef float v2f __attribute__((ext_vector_type(2)));
typedef float v8f __attribute__((ext_vector_type(8)));
typedef int   v4i __attribute__((vector_size(16)));   // matches builtin's expected pointee

#define AS1 __attribute__((address_space(1)))
#define AS3 __attribute__((address_space(3)))

// ---------------- data movement: global -> LDS (async when available) -------
__device__ __forceinline__ void copy16_to_lds(const float* __restrict__ g, float* l) {
#if __has_builtin(__builtin_amdgcn_global_load_async_to_lds_b128)
  __builtin_amdgcn_global_load_async_to_lds_b128((AS1 v4i*)g, (AS3 v4i*)l, 0, 0);
#else
  *(float4*)l = *(const float4*)g;
#endif
}

__device__ __forceinline__ void wait_lds_copies() {
#if __has_builtin(__builtin_amdgcn_global_load_async_to_lds_b128)
 #if __has_builtin(__builtin_amdgcn_s_wait_asynccnt)
  __builtin_amdgcn_s_wait_asynccnt(0);
 #else
  asm volatile("s_wait_asynccnt 0" ::: "memory");
 #endif
#else
  asm volatile("" ::: "memory");
#endif
}

__device__ __forceinline__ void wait_ds() {
  asm volatile("s_wait_dscnt 0" ::: "memory");
}

// ---------------- WMMA f32 16x16x4: C[M][N] += sum_k A[M][k]*B[k][N] --------
// A (16x4) and B (4x16) have the identical lane layout for "16 row-vectors x
// 4-K-chunk": lane L holds slot (L&15), k = (L>=16 ? 2:0) + {0,1}.
__device__ __forceinline__ v8f wmma4(v2f a, v2f b, v8f c) {
#if __has_builtin(__builtin_amdgcn_wmma_f32_16x16x4_f32)
  return __builtin_amdgcn_wmma_f32_16x16x4_f32(false, a, false, b, (short)0, c, false, false);
#else
  // functional fallback via cross-lane shuffles (compile-safety only)
  const int lane = threadIdx.x & 31;
  const int mb = (lane >= 16) ? 8 : 0;
  const int n  = lane & 15;
  #pragma unroll
  for (int v = 0; v < 8; ++v) {
    const int M = mb + v;
    float acc = c[v];
    #pragma unroll
    for (int k = 0; k < 4; ++k) {
      const float av = __shfl((k & 1) ? a.y : a.x, M + ((k >= 2) ? 16 : 0), 32);
      const float bv = __shfl((k & 1) ? b.y : b.x, n + ((k >= 2) ? 16 : 0), 32);
      acc += av * bv;
    }
    c[v] = acc;
  }
  return c;
#endif
}

__device__ __forceinline__ float pick8(v8f v, int i) {
  float r = v[0];
  #pragma unroll
  for (int t = 1; t < 8; ++t) r = (i == t) ? v[t] : r;
  return r;
}

__device__ __forceinline__ float wave_sum(float x) {
  #pragma unroll
  for (int o = 16; o > 0; o >>= 1) x += __shfl_xor(x, o, 32);
  return x;
}

__device__ __forceinline__ float one_minus_selfcos(float q) {
  const float s = fmaxf(sqrtf(fmaxf(q, 0.f)), EPSF);
  return 1.f - q / (s * s);
}

// ============ K1: per-row segment Gram (orthogonal + temporal + sumsq) ======
// One wave handles 4 pos rows -> 16 slots (row, segment). Gram = diag 4x4
// blocks of the 16x16 WMMA accumulator over K = 512.
__global__ __launch_bounds__(256) void k1_gram(const float* __restrict__ z,
                                               float* __restrict__ ss,
                                               float* __restrict__ part) {
  __shared__ __align__(16) float abufs[K1_WAVES * 16 * LSTR];
  __shared__ float cbufs[K1_WAVES * 256];
  __shared__ float red[K1_WAVES * 2];

  const int tid = threadIdx.x, wave = tid >> 5, lane = tid & 31;
  const int grp = blockIdx.x * K1_WAVES + wave;   // 4-row group, 4096 total
  const int r0  = grp * 4;
  float* ab = abufs + wave * 16 * LSTR;

  const int h    = lane >> 4;
  const int c4   = (lane & 15) * 4;
  const int s_op = lane & 15;
  const int kb   = h * 2;

  v8f c = {0.f, 0.f, 0.f, 0.f, 0.f, 0.f, 0.f, 0.f};

  for (int d0 = 0; d0 < TD; d0 += CHUNK) {
    wait_ds();   // previous chunk's operand reads done before overwrite
    #pragma unroll
    for (int j = 0; j < 8; ++j) {
      const int s = 2 * j + h;                      // slot = (row s>>2, seg s&3)
      const float* gp = z + (size_t)(r0 + (s >> 2)) * DIM + (s & 3) * TD + d0 + c4;
      copy16_to_lds(gp, ab + s * LSTR + c4);
    }
    wait_lds_copies();
    #pragma unroll
    for (int kk = 0; kk < CHUNK; kk += 4) {
      const v2f x = *(const v2f*)(ab + s_op * LSTR + kk + kb);
      c = wmma4(x, x, c);                           // self-Gram of the 16 slots
    }
  }

  // spill 16x16 accumulator to LDS: C[M][N], lane holds N=lane&15, M=v+(lane>=16?8:0)
  float* cb = cbufs + wave * 256;
  {
    const int mb = (lane >= 16) ? 8 : 0;
    const int n  = lane & 15;
    #pragma unroll
    for (int v = 0; v < 8; ++v) cb[(mb + v) * 16 + n] = c[v];
  }
  wait_ds();

  float orth = 0.f, temp = 0.f;
  if (lane < 4) {
    const int m = lane;
    #define G(a, b) cb[(4 * m + (a)) * 16 + (4 * m + (b))]
    const float g00 = G(0, 0), g11 = G(1, 1), g22 = G(2, 2), g33 = G(3, 3);
    const float n0 = fmaxf(sqrtf(fmaxf(g00, 0.f)), EPSF);
    const float n1 = fmaxf(sqrtf(fmaxf(g11, 0.f)), EPSF);
    const float n2 = fmaxf(sqrtf(fmaxf(g22, 0.f)), EPSF);
    const float n3 = fmaxf(sqrtf(fmaxf(g33, 0.f)), EPSF);
    orth = (fabsf(G(0, 1)) / (n0 * n1) +
            fabsf(G(1, 2)) / (n1 * n2) +
            fabsf(G(2, 3)) / (n2 * n3)) * (1.f / 3.f);
    const float q1 = g00 + g33 - 2.f * G(0, 3);     // ||z3-z0||^2
    const float q2 = g00 + g22 - 2.f * G(0, 2);     // ||z2-z0||^2
    const float q3 = g11 + g33 - 2.f * G(1, 3);     // ||z3-z1||^2
    temp = (one_minus_selfcos(q1) + one_minus_selfcos(q2) + one_minus_selfcos(q3)) * (1.f / 3.f);
    ss[r0 + m] = g00 + g11 + g22 + g33;             // full-row sumsq for K2
    #undef G
  }
  orth = wave_sum(orth);
  temp = wave_sum(temp);
  if (lane == 0) { red[wave * 2] = orth; red[wave * 2 + 1] = temp; }
  __syncthreads();
  if (tid == 0) {
    float o = 0.f, t = 0.f;
    for (int w = 0; w < K1_WAVES; ++w) { o += red[2 * w]; t += red[2 * w + 1]; }
    part[blockIdx.x * 2] = o;
    part[blockIdx.x * 2 + 1] = t;
  }
}

// ============ K2: align dots (pos·perm, pos·neg, neg·neg) via WMMA ==========
// One wave handles 16 pos rows; diagonals of the 16x16 accumulators are the
// needed per-row dots; pos/perm norms gathered from K1's sumsq array.
__global__ __launch_bounds__(256) void k2_align(const float* __restrict__ z,
                                                const int* __restrict__ perm,
                                                const float* __restrict__ ss,
                                                float* __restrict__ part) {
  __shared__ __align__(16) float bufs[K2_WAVES * 3 * 16 * LSTR];
  __shared__ float red[K2_WAVES * 2];

  const int tid = threadIdx.x, wave = tid >> 5, lane = tid & 31;
  const int tile = blockIdx.x * K2_WAVES + wave;   // 0..1023
  const int r0   = tile * 16;
  float* bp = bufs + wave * 3 * 16 * LSTR;  // pos rows
  float* bq = bp + 16 * LSTR;               // perm-gathered rows
  float* bn = bq + 16 * LSTR;               // neg rows

  const int h    = lane >> 4;
  const int c4   = (lane & 15) * 4;
  const int s_op = lane & 15;
  const int kb   = h * 2;

  int rows[8], prows[8];
  #pragma unroll
  for (int j = 0; j < 8; ++j) {
    const int r = 2 * j + h;
    rows[j]  = r0 + r;
    prows[j] = perm[r0 + r];
  }

  v8f c1 = {0.f,0.f,0.f,0.f,0.f,0.f,0.f,0.f};   // pos · perm^T
  v8f c2 = c1;                                  // pos · neg^T
  v8f c3 = c1;                                  // neg · neg^T

  for (int k0 = 0; k0 < DIM; k0 += CHUNK) {
    wait_ds();
    #pragma unroll
    for (int j = 0; j < 8; ++j) {
      const int r = 2 * j + h;
      copy16_to_lds(z + (size_t)rows[j] * DIM + k0 + c4,          bp + r * LSTR + c4);
      copy16_to_lds(z + (size_t)prows[j] * DIM + k0 + c4,         bq + r * LSTR + c4);
      copy16_to_lds(z + (size_t)(NPOS + rows[j]) * DIM + k0 + c4, bn + r * LSTR + c4);
    }
    wait_lds_copies();
    #pragma unroll
    for (int kk = 0; kk < CHUNK; kk += 4) {
      const int off = s_op * LSTR + kk + kb;
      const v2f xp = *(const v2f*)(bp + off);
      const v2f xq = *(const v2f*)(bq + off);
      const v2f xn = *(const v2f*)(bn + off);
      c1 = wmma4(xp, xq, c1);
      c2 = wmma4(xp, xn, c2);
      c3 = wmma4(xn, xn, c3);
    }
  }

  // diagonal owners: lanes 0-7 hold m=lane at c[lane]; lanes 24-31 hold m=lane-16 at c[lane-24]
  float* dg = bp;  // reuse staging LDS: [0..15]=dot_perm, [16..31]=dot_neg, [32..47]=ss_neg
  {
    int mv = -1; float d1 = 0.f, d2 = 0.f, d3 = 0.f;
    if (lane < 8)        { mv = lane;      d1 = pick8(c1, lane);      d2 = pick8(c2, lane);      d3 = pick8(c3, lane); }
    else if (lane >= 24) { mv = lane - 16; d1 = pick8(c1, lane - 24); d2 = pick8(c2, lane - 24); d3 = pick8(c3, lane - 24); }
    if (mv >= 0) { dg[mv] = d1; dg[16 + mv] = d2; dg[32 + mv] = d3; }
  }
  wait_ds();

  float cp = 0.f, cn = 0.f;
  if (lane < 16) {
    const int i = r0 + lane;
    const float ni = fmaxf(sqrtf(fmaxf(ss[i], 0.f)), EPSF);
    const float np = fmaxf(sqrtf(fmaxf(ss[perm[i]], 0.f)), EPSF);
    const float nn = fmaxf(sqrtf(fmaxf(dg[32 + lane], 0.f)), EPSF);
    cp = dg[lane]      / (ni * np);
    cn = dg[16 + lane] / (ni * nn);
  }
  cp = wave_sum(cp);
  cn = wave_sum(cn);
  if (lane == 0) { red[wave * 2] = cp; red[wave * 2 + 1] = cn; }
  __syncthreads();
  if (tid == 0) {
    float a = 0.f, b = 0.f;
    for (int w = 0; w < K2_WAVES; ++w) { a += red[2 * w]; b += red[2 * w + 1]; }
    part[blockIdx.x * 2] = a;
    part[blockIdx.x * 2 + 1] = b;
  }
}

// ============ K3: deterministic final fold ==================================
__global__ void k3_final(const float* __restrict__ p1, const float* __restrict__ p2,
                         float* __restrict__ out) {
  if (threadIdx.x == 0 && blockIdx.x == 0) {
    float o = 0.f, t = 0.f, cp = 0.f, cn = 0.f;
    for (int i = 0; i < K1_BLOCKS; ++i) { o  += p1[2 * i]; t  += p1[2 * i + 1]; }
    for (int i = 0; i < K2_BLOCKS; ++i) { cp += p2[2 * i]; cn += p2[2 * i + 1]; }
    const float inv = 1.0f / (float)NPOS;
    out[0] = (1.f - cp * inv) + cn * inv;  // align
    out[1] = o * inv;                      // orthogonal
    out[2] = t * inv;                      // temporal
  }
}

extern "C" void kernel_launch(void* const* d_in, const int* in_sizes, int n_in,
                              void* d_out, int out_size, void* d_ws, size_t ws_size,
                              hipStream_t stream) {
  (void)in_sizes; (void)n_in; (void)out_size; (void)ws_size;
  const float* z   = (const float*)d_in[0];
  // d_in[1] = labels: structurally known (first half positive), unused.
  const int* perm  = (const int*)d_in[2];

  float* ws = (float*)d_ws;
  float* ss = ws;                        // NPOS floats: per-pos-row sumsq
  float* p1 = ss + NPOS;                 // K1_BLOCKS*2 partials (orth, temporal)
  float* p2 = p1 + K1_BLOCKS * 2;        // K2_BLOCKS*2 partials (cos_perm, cos_neg)

  k1_gram <<<K1_BLOCKS, 256, 0, stream>>>(z, ss, p1);
  k2_align<<<K2_BLOCKS, 256, 0, stream>>>(z, perm, ss, p2);
  k3_final<<<1, 32, 0, stream>>>(p1, p2, (float*)d_out);
}